// OnlineLabelSmoothLoss_64132451664542
// MI455X (gfx1250) — compile-verified
//
#include <hip/hip_runtime.h>
#include <math.h>

typedef __attribute__((ext_vector_type(2))) float v2f;
typedef __attribute__((ext_vector_type(4))) float v4f;
typedef __attribute__((ext_vector_type(8))) float v8f;

#define BB     8192
#define CC     8192
#define ROWS   16           // rows per workgroup
#define TILE_K 128          // columns per LDS tile
#define NTILE  (CC / TILE_K)
#define NTHR   256          // 8 waves (wave32)
#define NWAVE  8
#define LDSTR  (TILE_K + 4) // padded row stride (floats): 16B align kept, banks spread

// ---------------------------------------------------------------------------
// Zero-init output: [0]=loss, [1 .. 1+C*C)=soft_labels_update, then C counts
// ---------------------------------------------------------------------------
__global__ __launch_bounds__(256) void ols_zero(float* __restrict__ out, long long n) {
    const long long n4 = n >> 2;
    const v4f z = {0.f, 0.f, 0.f, 0.f};
    v4f* o4 = (v4f*)out;
    const long long stride = (long long)gridDim.x * blockDim.x;
    for (long long i = (long long)blockIdx.x * blockDim.x + threadIdx.x; i < n4; i += stride)
        __builtin_nontemporal_store(z, &o4[i]);   // don't pollute L2 with zeros
    if (blockIdx.x == 0 && threadIdx.x == 0)
        for (long long i = n4 << 2; i < n; ++i) out[i] = 0.0f;
}

// ---------------------------------------------------------------------------
// Fused: pipelined tile stream + branchless online softmax + WMMA row-dots
// ---------------------------------------------------------------------------
__global__ __launch_bounds__(NTHR) void ols_main(const float* __restrict__ input,
                                                 const int*   __restrict__ target,
                                                 const float* __restrict__ soft_labels,
                                                 float*       __restrict__ out) {
    __shared__ float in_tile[2][ROWS][LDSTR];
    __shared__ float sl_tile[2][ROWS][LDSTR];
    __shared__ float red_m[ROWS][16];
    __shared__ float red_s[ROWS][16];
    __shared__ int   red_i[ROWS][16];
    __shared__ float dot_red[NWAVE][ROWS];
    __shared__ int   t_sh[ROWS];
    __shared__ float row_M[ROWS];
    __shared__ float row_invS[ROWS];
    __shared__ int   row_flag[ROWS];

    const int tid   = threadIdx.x;
    const int lane  = tid & 31;
    const int wave  = tid >> 5;
    const int rbase = blockIdx.x * ROWS;

    if (tid < ROWS) {
        t_sh[tid]     = target[rbase + tid];
        row_flag[tid] = 0;
    }
    __syncthreads();

    // ---- staging role: thread feeds rows lr and lr+8 at column cst (16B each)
    const int lr  = tid >> 5;            // 0..7
    const int cst = (tid & 31) << 2;     // 0,4,...,124
    const int t0  = t_sh[lr];
    const int t1  = t_sh[lr + 8];
    const float* gin0 = input + (size_t)(rbase + lr)     * CC + cst;
    const float* gin1 = input + (size_t)(rbase + lr + 8) * CC + cst;
    const float* gsl0 = soft_labels + (size_t)t0 * CC + cst;
    const float* gsl1 = soft_labels + (size_t)t1 * CC + cst;

    // ---- stats role: row srow, strip [ssub*8, ssub*8+8) of every tile
    const int srow = tid >> 4;
    const int ssub = tid & 15;
    float m = -INFINITY, s = 0.0f;
    int   am = 0x7fffffff;

    // ---- WMMA fragment role (ISA 7.12.2 f32 16x4 A-layout):
    // lanes 0-15 carry K={0,1}, lanes 16-31 carry K={2,3}; matrix row = lane%16
    const int frow = lane & 15;
    const int fsel = (lane >> 4) << 1;   // 0 or 2
    // single shared-base pointers; buffer select folded into integer offset so
    // addrspace(3) inference survives -> ds_load (not flat_load)
    const float* finp  = &in_tile[0][frow][0];
    const float* fslp  = &sl_tile[0][frow][0];
    const float* statp = &in_tile[0][srow][ssub << 3];
    const int    BUFO  = ROWS * LDSTR;   // floats per buffer
    v8f acc = {};                        // 16x16 f32 accumulator; diag = row dots

    // ---- prologue: stage tile 0 into buffer 0
    {
        v4f a0 = __builtin_nontemporal_load((const v4f*)gin0);
        v4f a1 = __builtin_nontemporal_load((const v4f*)gin1);
        v4f b0 = *(const v4f*)gsl0;
        v4f b1 = *(const v4f*)gsl1;
        *(v4f*)&in_tile[0][lr][cst]     = a0;
        *(v4f*)&in_tile[0][lr + 8][cst] = a1;
        *(v4f*)&sl_tile[0][lr][cst]     = b0;
        *(v4f*)&sl_tile[0][lr + 8][cst] = b1;
    }
    __syncthreads();

    for (int it = 0; it < NTILE; ++it) {
        const int  cur  = it & 1;
        const int  nxt  = cur ^ 1;
        const bool more = (it + 1) < NTILE;
        const int  boff = cur * BUFO;

        // --- issue next tile's global loads (latency hidden by compute below)
        v4f a0, a1, b0, b1;
        if (more) {
            const int off = (it + 1) * TILE_K;
            a0 = __builtin_nontemporal_load((const v4f*)(gin0 + off)); // read-once stream
            a1 = __builtin_nontemporal_load((const v4f*)(gin1 + off));
            b0 = *(const v4f*)(gsl0 + off);   // gathered rows: keep L2-temporal
            b1 = *(const v4f*)(gsl1 + off);
            if (it + 2 < NTILE) {   // global_prefetch_b8 two tiles ahead
                __builtin_prefetch(gsl0 + (it + 2) * TILE_K, 0, 1);
                __builtin_prefetch(gsl1 + (it + 2) * TILE_K, 0, 1);
            }
        }

        // --- branchless online softmax over this thread's 8-element strip
        {
            const float* rowp = statp + boff;
            const v4f xa = *(const v4f*)rowp;
            const v4f xb = *(const v4f*)(rowp + 4);
            const int cb = it * TILE_K + (ssub << 3);
            float mv = m; int amv = am;
            { const bool g = xa.x > mv; amv = g ? cb + 0 : amv; mv = g ? xa.x : mv; }
            { const bool g = xa.y > mv; amv = g ? cb + 1 : amv; mv = g ? xa.y : mv; }
            { const bool g = xa.z > mv; amv = g ? cb + 2 : amv; mv = g ? xa.z : mv; }
            { const bool g = xa.w > mv; amv = g ? cb + 3 : amv; mv = g ? xa.w : mv; }
            { const bool g = xb.x > mv; amv = g ? cb + 4 : amv; mv = g ? xb.x : mv; }
            { const bool g = xb.y > mv; amv = g ? cb + 5 : amv; mv = g ? xb.y : mv; }
            { const bool g = xb.z > mv; amv = g ? cb + 6 : amv; mv = g ? xb.z : mv; }
            { const bool g = xb.w > mv; amv = g ? cb + 7 : amv; mv = g ? xb.w : mv; }
            const float scale = __expf(m - mv);   // exp(-inf)=0 handles first tile
            const float ssum =
                __expf(xa.x - mv) + __expf(xa.y - mv) + __expf(xa.z - mv) + __expf(xa.w - mv) +
                __expf(xb.x - mv) + __expf(xb.y - mv) + __expf(xb.z - mv) + __expf(xb.w - mv);
            s  = s * scale + ssum;
            m  = mv;
            am = amv;
        }

        // --- WMMA: wave w covers tile cols [w*16, w*16+16), K-steps of 4.
        // acc += A(16x4 input rows) * B(4x16 gathered soft-label rows)
        #pragma unroll
        for (int j = 0; j < 4; ++j) {
            const int koff = boff + (wave << 4) + (j << 2) + fsel;
            v2f A  = *(const v2f*)(finp + koff);
            v2f Bm = *(const v2f*)(fslp + koff);
            acc = __builtin_amdgcn_wmma_f32_16x16x4_f32(
                false, A, false, Bm, (short)0, acc, false, false);
        }

        // --- drain staged registers into the other buffer
        if (more) {
            *(v4f*)&in_tile[nxt][lr][cst]     = a0;
            *(v4f*)&in_tile[nxt][lr + 8][cst] = a1;
            *(v4f*)&sl_tile[nxt][lr][cst]     = b0;
            *(v4f*)&sl_tile[nxt][lr + 8][cst] = b1;
        }
        __syncthreads();
    }

    // --- accumulator diagonal: D[m,m] for m<8 at (vgpr m, lane m);
    //     for m>=8 at (vgpr m-8, lane m+16)
    #pragma unroll
    for (int v = 0; v < 8; ++v) {
        if (lane == v)      dot_red[wave][v]     = acc[v];
        if (lane == v + 24) dot_red[wave][v + 8] = acc[v];
    }
    red_m[srow][ssub] = m;
    red_s[srow][ssub] = s;
    red_i[srow][ssub] = am;
    __syncthreads();

    // --- per-row combine (threads 0..15)
    if (tid < ROWS) {
        const int r = rbase + tid;
        float M = -INFINITY;
        #pragma unroll
        for (int i = 0; i < 16; ++i) M = fmaxf(M, red_m[tid][i]);
        float S = 0.0f; int idx = 0x7fffffff;
        #pragma unroll
        for (int i = 0; i < 16; ++i) {
            S += red_s[tid][i] * __expf(red_m[tid][i] - M);
            if (red_m[tid][i] == M && red_i[tid][i] < idx) idx = red_i[tid][i];
        }
        float dot = 0.0f;
        #pragma unroll
        for (int w = 0; w < NWAVE; ++w) dot += dot_red[w][tid];

        const int   t  = t_sh[tid];
        const float xt = input[(size_t)r * CC + t];
        const float L  = __logf(S);
        // loss contrib: (1/B) * [ M + L - 0.5*dot - 0.5*x_t ]
        atomicAdd(&out[0], (M + L - 0.5f * (dot + xt)) * (1.0f / (float)BB));

        if (idx == t) {  // correct prediction (rare): flag row for prob scatter
            atomicAdd(&out[1 + (size_t)CC * CC + t], 1.0f);
            row_M[tid]    = M;
            row_invS[tid] = 1.0f / S;
            row_flag[tid] = 1;
        }
    }
    __syncthreads();

    // --- rare path: add softmax prob row into soft_labels_update[t,:]
    float* upd = out + 1;
    for (int mr = 0; mr < ROWS; ++mr) {
        if (row_flag[mr]) {
            const int   r   = rbase + mr;
            const int   t   = t_sh[mr];
            const float Mv  = row_M[mr];
            const float inv = row_invS[mr];
            for (int c = tid; c < CC; c += NTHR) {
                const float p = __expf(input[(size_t)r * CC + c] - Mv) * inv;
                atomicAdd(&upd[(size_t)t * CC + c], p);
            }
        }
    }
}

// ---------------------------------------------------------------------------
extern "C" void kernel_launch(void* const* d_in, const int* in_sizes, int n_in,
                              void* d_out, int out_size, void* d_ws, size_t ws_size,
                              hipStream_t stream) {
    const float* input       = (const float*)d_in[0];
    const int*   target      = (const int*)d_in[1];
    const float* soft_labels = (const float*)d_in[2];
    float*       out         = (float*)d_out;

    const long long n = 1LL + (long long)CC * CC + (long long)CC;
    ols_zero<<<2048, 256, 0, stream>>>(out, n);
    ols_main<<<BB / ROWS, NTHR, 0, stream>>>(input, target, soft_labels, out);
}